// AdptWeightBCEDiceLoss_75239237091946
// MI455X (gfx1250) — compile-verified
//
#include <hip/hip_runtime.h>
#include <hip/hip_bf16.h>

typedef __attribute__((ext_vector_type(2))) float v2f;
typedef __attribute__((ext_vector_type(8))) float v8f;

#define HW   512
#define IMG  (HW * HW)
#define NIMG 32
// LDS row stride in words: 16 halo + 512 data + 32 halo = 560, padded to 580
// (580 mod 64 == 4 -> conflict-free ds_load_b64 across 32 lanes)
#define LDS_STRIDE 580
#define HALO 16

#if __has_builtin(__builtin_amdgcn_global_load_async_to_lds_b128)
#define USE_ASYNC_LDS 1
typedef int v4i __attribute__((ext_vector_type(4)));
typedef __attribute__((address_space(1))) v4i g_v4i;   // global (AS1)
typedef __attribute__((address_space(3))) v4i l_v4i;   // LDS (AS3)
#endif

// -----------------------------------------------------------------------------
// Pass 1: horizontal 31-tap box sum H = T x Band via V_WMMA_F32_16X16X4_F32.
// One block = one image x one 16-row stripe. 8 waves x 4 column tiles each.
// Staging uses the CDNA5 async global->LDS path (ASYNCcnt) when available.
// -----------------------------------------------------------------------------
__global__ __launch_bounds__(256) void hpool_wmma_kernel(
    const float* __restrict__ T, float* __restrict__ H) {
  __shared__ float tile[16 * LDS_STRIDE];

  const int b  = blockIdx.x >> 5;          // image
  const int r0 = (blockIdx.x & 31) << 4;   // stripe base row
  const float* Timg = T + (size_t)b * IMG + (size_t)r0 * HW;

  // Stage 16x512 floats into LDS at column offset HALO.
#ifdef USE_ASYNC_LDS
  // 16B per lane per iteration, LDS address per lane -> maps 1:1 onto
  // GLOBAL_LOAD_ASYNC_TO_LDS_B128 (no VGPR round-trip, tracked by ASYNCcnt).
  for (int i = threadIdx.x; i < 16 * 128; i += 256) {
    const int row = i >> 7;
    const int c4  = (i & 127) << 2;
    __builtin_amdgcn_global_load_async_to_lds_b128(
        (g_v4i*)(Timg + row * HW + c4),
        (l_v4i*)&tile[row * LDS_STRIDE + HALO + c4],
        /*imm_offset=*/0, /*cpol=*/0);
  }
#else
  for (int i = threadIdx.x; i < 16 * 128; i += 256) {
    const int row = i >> 7;
    const int c4  = (i & 127) << 2;
    const float4 v = *(const float4*)(Timg + row * HW + c4);
    float* dst = &tile[row * LDS_STRIDE + HALO + c4];
    dst[0] = v.x; dst[1] = v.y; dst[2] = v.z; dst[3] = v.w;
  }
#endif
  // Zero halos: words [0,16) and [528,560) of each row (zero-pad == count_include_pad).
  for (int i = threadIdx.x; i < 16 * 48; i += 256) {
    const int row = i / 48;
    const int j   = i % 48;
    tile[row * LDS_STRIDE + ((j < 16) ? j : (HW + j))] = 0.0f;
  }
#ifdef USE_ASYNC_LDS
#if __has_builtin(__builtin_amdgcn_s_wait_asynccnt)
  __builtin_amdgcn_s_wait_asynccnt(0);
#else
  asm volatile("s_wait_asynccnt 0" ::: "memory");
#endif
#endif
  __syncthreads();

  const int wave = threadIdx.x >> 5;
  const int lane = threadIdx.x & 31;
  const int n    = lane & 15;   // output column within tile (B/D layout)
  const int m    = lane & 15;   // A-matrix row (lanes 0-15 and 16-31 both hold M=0..15)
  const int khi  = lane >> 4;   // selects K-pair {0,1} vs {2,3} within a k-step
  float* Himg = H + (size_t)b * IMG;

  for (int j = 0; j < 4; ++j) {
    const int c0 = (wave + (j << 3)) << 4;            // tile base column
    // A row in LDS: global k = c0-16+q  ->  word index = HALO + c0-16+q = c0+q
    const float* arow = &tile[m * LDS_STRIDE + c0];
    v8f acc = {};
#pragma unroll
    for (int s = 0; s < 12; ++s) {
      const int q = (s << 2) + (khi << 1);            // local k of this lane's VGPR0
      v2f a;
      a.x = arow[q];
      a.y = arow[q + 1];
      // Band matrix element: S[k][c]=1 iff (k_global - c_global) in [-15,15]
      //  -> q' - n in [1,31] with q' = q or q+1.
      const int d0 = q - n;
      v2f bv;
      bv.x = (d0 >= 1 && d0 <= 31) ? 1.0f : 0.0f;
      bv.y = (d0 >= 0 && d0 <= 30) ? 1.0f : 0.0f;
      acc = __builtin_amdgcn_wmma_f32_16x16x4_f32(
          /*neg_a=*/false, a, /*neg_b=*/false, bv,
          /*c_mod=*/(short)0, acc, /*reuse_a=*/false, /*reuse_b=*/false);
    }
    // D layout: VGPR v -> row r0+v (lanes 0-15) / r0+v+8 (lanes 16-31), col c0+n.
#pragma unroll
    for (int v = 0; v < 8; ++v) {
      Himg[(size_t)(r0 + v + (khi << 3)) * HW + c0 + n] = acc[v];
    }
  }
}

// -----------------------------------------------------------------------------
// Pass 2: vertical 31-row running sum over H (L2-resident) + fused elementwise
// loss terms + per-block partial reductions. One block = image x 64-row chunk,
// one thread per column.
// -----------------------------------------------------------------------------
__global__ __launch_bounds__(512) void fuse_kernel(
    const float* __restrict__ P, const float* __restrict__ T,
    const float* __restrict__ H, float* __restrict__ partials) {
  const int b  = blockIdx.x >> 3;
  const int r0 = (blockIdx.x & 7) << 6;   // 64-row chunk
  const int c  = threadIdx.x;
  const float* Himg = H + (size_t)b * IMG;
  const float* Timg = T + (size_t)b * IMG;
  const float* Pimg = P + (size_t)b * IMG;

  // Warm up the 31-row window for output row r0 (rows clipped; pad rows add 0).
  float s = 0.0f;
  const int lo = (r0 - 15 < 0) ? 0 : r0 - 15;
  const int hi = (r0 + 15 > HW - 1) ? HW - 1 : r0 + 15;
  for (int rr = lo; rr <= hi; ++rr) s += Himg[(size_t)rr * HW + c];

  float wsum = 0.0f, inter = 0.0f, uni = 0.0f, bce = 0.0f;
  for (int i = 0; i < 64; ++i) {
    const int r = r0 + i;
    if (((i & 15) == 0) && (r + 47 < HW))
      __builtin_prefetch(&Himg[(size_t)(r + 47) * HW + c], 0, 0);

    const float pool = s * (1.0f / 961.0f);           // 31*31, count_include_pad
    const float t = Timg[(size_t)r * HW + c];
    const float x = Pimg[(size_t)r * HW + c];
    const float w = 1.0f + 5.0f * fabsf(pool - t);

    // softplus(x) - x*t, numerically stable
    bce += fmaxf(x, 0.0f) + log1pf(expf(-fabsf(x))) - x * t;

    const float p = 1.0f / (1.0f + expf(-x));
    wsum  += w;
    inter += p * t * w;
    uni   += (p + t) * w;

    // Slide window r -> r+1
    const int ra = r + 16, rs = r - 15;
    if (ra < HW)  s += Himg[(size_t)ra * HW + c];
    if (rs >= 0)  s -= Himg[(size_t)rs * HW + c];
  }

  // Deterministic block tree reduction of 4 accumulators.
  __shared__ float red[512 * 4];
  const int t4 = threadIdx.x << 2;
  red[t4 + 0] = wsum; red[t4 + 1] = inter; red[t4 + 2] = uni; red[t4 + 3] = bce;
  __syncthreads();
  for (int off = 256; off >= 1; off >>= 1) {
    if (threadIdx.x < off) {
      const int o4 = (threadIdx.x + off) << 2;
      red[t4 + 0] += red[o4 + 0];
      red[t4 + 1] += red[o4 + 1];
      red[t4 + 2] += red[o4 + 2];
      red[t4 + 3] += red[o4 + 3];
    }
    __syncthreads();
  }
  if (threadIdx.x == 0) {
    float4 o;
    o.x = red[0]; o.y = red[1]; o.z = red[2]; o.w = red[3];
    ((float4*)partials)[blockIdx.x] = o;
  }
}

// -----------------------------------------------------------------------------
// Pass 3: combine 256 block partials -> scalar loss.
// -----------------------------------------------------------------------------
__global__ __launch_bounds__(256) void finalize_kernel(
    const float* __restrict__ partials, float* __restrict__ out) {
  __shared__ float sh[256 * 4];
  __shared__ float img[NIMG * 4];
  const int t = threadIdx.x;
  const float4 v = ((const float4*)partials)[t];
  sh[t * 4 + 0] = v.x; sh[t * 4 + 1] = v.y; sh[t * 4 + 2] = v.z; sh[t * 4 + 3] = v.w;
  __syncthreads();
  if (t < NIMG) {
    float s0 = 0, s1 = 0, s2 = 0, s3 = 0;
    for (int e = t * 8; e < t * 8 + 8; ++e) {
      s0 += sh[e * 4 + 0]; s1 += sh[e * 4 + 1];
      s2 += sh[e * 4 + 2]; s3 += sh[e * 4 + 3];
    }
    img[t * 4 + 0] = s0; img[t * 4 + 1] = s1; img[t * 4 + 2] = s2; img[t * 4 + 3] = s3;
  }
  __syncthreads();
  if (t == 0) {
    const float SMOOTH = 1e-8f;
    float bce_tot = 0.0f;
    for (int bb = 0; bb < NIMG; ++bb) bce_tot += img[bb * 4 + 3];
    const float bce = bce_tot / (float)((size_t)NIMG * IMG);
    float acc = 0.0f;
    for (int bb = 0; bb < NIMG; ++bb) {
      const float wsum  = img[bb * 4 + 0];
      const float inter = img[bb * 4 + 1];
      const float uni   = img[bb * 4 + 2];
      const float wbce = (wsum * bce + SMOOTH) / (wsum + SMOOTH);
      const float wiou = 1.0f - (inter + 1.0f + SMOOTH) / (uni - inter + 1.0f + SMOOTH);
      acc += wbce + wiou;
    }
    out[0] = acc / (float)NIMG;
  }
}

extern "C" void kernel_launch(void* const* d_in, const int* in_sizes, int n_in,
                              void* d_out, int out_size, void* d_ws, size_t ws_size,
                              hipStream_t stream) {
  const float* y_pred   = (const float*)d_in[0];
  const float* y_target = (const float*)d_in[1];
  float* H        = (float*)d_ws;                                   // 32 MiB
  float* partials = (float*)((char*)d_ws + (size_t)NIMG * IMG * sizeof(float));

  hpool_wmma_kernel<<<dim3(NIMG * 32), dim3(256), 0, stream>>>(y_target, H);
  fuse_kernel<<<dim3(NIMG * 8), dim3(512), 0, stream>>>(y_pred, y_target, H, partials);
  finalize_kernel<<<dim3(1), dim3(256), 0, stream>>>(partials, (float*)d_out);
}